// MambaBlock_8083128451705
// MI455X (gfx1250) — compile-verified
//
#include <hip/hip_runtime.h>
#include <hip/hip_bf16.h>
#include <math.h>

// ---------------------------------------------------------------------------
// Mamba block for MI455X (gfx1250), wave32, fp32 WMMA path.
//   B=4, T=2048, D_MODEL=512, D_INNER=1024, D_STATE=16, D_CONV=4
// ---------------------------------------------------------------------------

#define BB      4
#define TT      2048
#define DMODEL  512
#define DINNER  1024
#define DSTATE  16
#define DCONV   4
#define MROWS   (BB * TT)          // 8192

typedef __attribute__((ext_vector_type(2))) float v2f;
typedef __attribute__((ext_vector_type(8))) float v8f;

__device__ __forceinline__ float softplus_f(float v) {
    return (v > 20.0f) ? v : log1pf(expf(v));
}

// ---------------------------------------------------------------------------
// 32x32-per-wave fp32 WMMA GEMM:  C[M x N] = A[M x K] * W[N x K]^T (+epilogue)
// One wave computes a 2x2 grid of 16x16 WMMA tiles -> 4 independent
// accumulator chains (hides XDL latency) and 2x fragment reuse
// (4 loads -> 4 WMMAs per k-step).
//
// Per CDNA5 ISA 7.12.2 (32-bit A 16x4): lanes 0-15 hold (K=0,K=1) for row
// M=lane, lanes 16-31 hold (K=2,K=3). B (4x16) mirrors with N=lane&15.
// C/D: VGPR i holds M=i (lanes 0-15) / M=8+i (lanes 16-31), N=lane&15.
//
// Requires M % 32 == 0, N % 32 == 0, K % 4 == 0.
// mode 0: plain store;  mode 1: v = softplus(v + bias[n])  (dt_proj)
// ---------------------------------------------------------------------------
__global__ __launch_bounds__(32) void gemm_wmma_f32_32x32(
    const float* __restrict__ A, const float* __restrict__ W,
    const float* __restrict__ bias, float* __restrict__ C,
    int M, int N, int K, int mode)
{
    const int tile_n = blockIdx.x << 5;
    const int tile_m = blockIdx.y << 5;
    const int lane   = threadIdx.x;         // 0..31
    const int r      = lane & 15;
    const int kb     = (lane >> 4) << 1;    // 0 or 2

    const float* __restrict__ a0p = A + (size_t)(tile_m + r) * K + kb;
    const float* __restrict__ a1p = a0p + (size_t)16 * K;
    const float* __restrict__ b0p = W + (size_t)(tile_n + r) * K + kb;
    const float* __restrict__ b1p = b0p + (size_t)16 * K;

    v8f acc00 = {}, acc01 = {}, acc10 = {}, acc11 = {};

    #pragma unroll 2
    for (int k0 = 0; k0 < K; k0 += 4) {
        v2f a0 = *(const v2f*)(a0p + k0);
        v2f a1 = *(const v2f*)(a1p + k0);
        v2f b0 = *(const v2f*)(b0p + k0);
        v2f b1 = *(const v2f*)(b1p + k0);
        acc00 = __builtin_amdgcn_wmma_f32_16x16x4_f32(false, a0, false, b0,
                                                      (short)0, acc00, false, false);
        acc01 = __builtin_amdgcn_wmma_f32_16x16x4_f32(false, a0, false, b1,
                                                      (short)0, acc01, false, false);
        acc10 = __builtin_amdgcn_wmma_f32_16x16x4_f32(false, a1, false, b0,
                                                      (short)0, acc10, false, false);
        acc11 = __builtin_amdgcn_wmma_f32_16x16x4_f32(false, a1, false, b1,
                                                      (short)0, acc11, false, false);
    }

    const int col   = lane & 15;
    const int mbase = (lane >> 4) << 3;     // 0 or 8

    #pragma unroll
    for (int j = 0; j < 2; ++j) {
        const int n = tile_n + (j << 4) + col;
        const float bval = (mode == 1) ? bias[n] : 0.0f;
        #pragma unroll
        for (int i = 0; i < 8; ++i) {
            const int m0 = tile_m + mbase + i;
            float v0 = (j == 0) ? acc00[i] : acc01[i];
            float v1 = (j == 0) ? acc10[i] : acc11[i];
            if (mode == 1) { v0 = softplus_f(v0 + bval); v1 = softplus_f(v1 + bval); }
            C[(size_t)m0 * N + n]        = v0;
            C[(size_t)(m0 + 16) * N + n] = v1;
        }
    }
}

// ---------------------------------------------------------------------------
// 16x16-per-wave fp32 WMMA GEMM for narrow N (b_proj / c_proj, N = 16).
// ---------------------------------------------------------------------------
__global__ __launch_bounds__(32) void gemm_wmma_f32_16(
    const float* __restrict__ A, const float* __restrict__ W,
    float* __restrict__ C, int M, int N, int K)
{
    const int tile_n = blockIdx.x << 4;
    const int tile_m = blockIdx.y << 4;
    const int lane   = threadIdx.x;
    const int r      = lane & 15;
    const int kb     = (lane >> 4) << 1;

    const float* __restrict__ arow = A + (size_t)(tile_m + r) * K + kb;
    const float* __restrict__ brow = W + (size_t)(tile_n + r) * K + kb;

    v8f acc = {};
    #pragma unroll 4
    for (int k0 = 0; k0 < K; k0 += 4) {
        v2f a = *(const v2f*)(arow + k0);
        v2f b = *(const v2f*)(brow + k0);
        acc = __builtin_amdgcn_wmma_f32_16x16x4_f32(false, a, false, b,
                                                    (short)0, acc, false, false);
    }

    const int col   = lane & 15;
    const int mbase = (lane >> 4) << 3;
    #pragma unroll
    for (int i = 0; i < 8; ++i)
        C[(size_t)(tile_m + mbase + i) * N + tile_n + col] = acc[i];
}

// ---------------------------------------------------------------------------
// Causal depthwise conv1d (kernel 4, left pad 3) + bias + SiLU.
// Input xs is the first DINNER columns of xr (row stride 2*DINNER).
// ---------------------------------------------------------------------------
__global__ __launch_bounds__(256) void conv_silu_kernel(
    const float* __restrict__ xr, const float* __restrict__ cw,
    const float* __restrict__ cb, float* __restrict__ xc)
{
    const int idx = blockIdx.x * 256 + threadIdx.x;   // over B*T*DINNER
    const int d   = idx & (DINNER - 1);
    const int bt  = idx >> 10;                        // DINNER = 1024
    const int t   = bt & (TT - 1);
    const int b   = bt >> 11;                         // TT = 2048

    const float* __restrict__ w = cw + d * DCONV;
    float acc = cb[d];
    #pragma unroll
    for (int k = 0; k < DCONV; ++k) {
        const int tt = t - (DCONV - 1) + k;
        if (tt >= 0)
            acc += xr[(size_t)(b * TT + tt) * (2 * DINNER) + d] * w[k];
    }
    xc[idx] = acc * (1.0f / (1.0f + expf(-acc)));     // SiLU
}

// ---------------------------------------------------------------------------
// Selective scan. One thread per (b, d) channel; h[DSTATE] lives in VGPRs.
// Bm/Cm addresses depend only on (b, t) -> wave-uniform scalar path.
// Fused: y += x*D, gating by SiLU(res), write yg.
// Prefetch next timestep's delta/xc rows (gfx1250 global_prefetch_b8).
// ---------------------------------------------------------------------------
__global__ __launch_bounds__(256) void scan_kernel(
    const float* __restrict__ delta, const float* __restrict__ Bm,
    const float* __restrict__ Cm,    const float* __restrict__ xc,
    const float* __restrict__ xr,    const float* __restrict__ A_log,
    const float* __restrict__ Dp,    float* __restrict__ yg)
{
    const int gid = blockIdx.x * 256 + threadIdx.x;   // 0 .. B*DINNER-1
    const int b   = gid >> 10;                        // DINNER = 1024
    const int d   = gid & (DINNER - 1);

    float Areg[DSTATE], h[DSTATE];
    #pragma unroll
    for (int n = 0; n < DSTATE; ++n) {
        Areg[n] = -expf(A_log[d * DSTATE + n]);
        h[n]    = 0.0f;
    }
    const float Dd = Dp[d];

    for (int t = 0; t < TT; ++t) {
        const int row = b * TT + t;
        const float dt = delta[(size_t)row * DINNER + d];
        const float xv = xc[(size_t)row * DINNER + d];
        const float* __restrict__ Bt = Bm + (size_t)row * DSTATE;
        const float* __restrict__ Ct = Cm + (size_t)row * DSTATE;

        if (t + 1 < TT) {   // pull next step's rows toward the WGP
            __builtin_prefetch(delta + (size_t)(row + 1) * DINNER + d, 0, 1);
            __builtin_prefetch(xc    + (size_t)(row + 1) * DINNER + d, 0, 1);
        }

        const float dx = dt * xv;
        float y = 0.0f;
        #pragma unroll
        for (int n = 0; n < DSTATE; ++n) {
            const float dA = expf(dt * Areg[n]);
            h[n] = dA * h[n] + dx * Bt[n];
            y += h[n] * Ct[n];
        }
        y += xv * Dd;

        const float res = xr[(size_t)row * (2 * DINNER) + DINNER + d];
        const float g   = res * (1.0f / (1.0f + expf(-res)));   // SiLU(res)
        yg[(size_t)row * DINNER + d] = y * g;
    }
}

// ---------------------------------------------------------------------------
// Launch
// ---------------------------------------------------------------------------
extern "C" void kernel_launch(void* const* d_in, const int* in_sizes, int n_in,
                              void* d_out, int out_size, void* d_ws, size_t ws_size,
                              hipStream_t stream)
{
    const float* x          = (const float*)d_in[0];
    const float* in_proj_w  = (const float*)d_in[1];
    const float* conv_w     = (const float*)d_in[2];
    const float* conv_b     = (const float*)d_in[3];
    const float* b_proj_w   = (const float*)d_in[4];
    const float* c_proj_w   = (const float*)d_in[5];
    const float* dt_proj_w  = (const float*)d_in[6];
    const float* dt_proj_b  = (const float*)d_in[7];
    const float* A_log      = (const float*)d_in[8];
    const float* Dp         = (const float*)d_in[9];
    const float* out_proj_w = (const float*)d_in[10];
    float* out = (float*)d_out;

    float* xr    = (float*)d_ws;                         // 8192 x 2048
    float* xc    = xr    + (size_t)MROWS * 2 * DINNER;   // 8192 x 1024
    float* delta = xc    + (size_t)MROWS * DINNER;       // 8192 x 1024
    float* Bm    = delta + (size_t)MROWS * DINNER;       // 8192 x 16
    float* Cm    = Bm    + (size_t)MROWS * DSTATE;       // 8192 x 16
    float* yg    = Cm    + (size_t)MROWS * DSTATE;       // 8192 x 1024

    const dim3 wave(32);

    // 1) in_proj:  xr = x @ in_proj_w^T      (8192 x 2048, K=512)
    gemm_wmma_f32_32x32<<<dim3((2 * DINNER) / 32, MROWS / 32), wave, 0, stream>>>(
        x, in_proj_w, nullptr, xr, MROWS, 2 * DINNER, DMODEL, 0);

    // 2) causal depthwise conv + SiLU -> xc
    conv_silu_kernel<<<(MROWS * DINNER) / 256, 256, 0, stream>>>(
        xr, conv_w, conv_b, xc);

    // 3) delta = softplus(xc @ dt_proj_w^T + b)   (8192 x 1024, K=1024)
    gemm_wmma_f32_32x32<<<dim3(DINNER / 32, MROWS / 32), wave, 0, stream>>>(
        xc, dt_proj_w, dt_proj_b, delta, MROWS, DINNER, DINNER, 1);

    // 4) Bm = xc @ b_proj_w^T, Cm = xc @ c_proj_w^T   (8192 x 16, K=1024)
    gemm_wmma_f32_16<<<dim3(1, MROWS / 16), wave, 0, stream>>>(
        xc, b_proj_w, Bm, MROWS, DSTATE, DINNER);
    gemm_wmma_f32_16<<<dim3(1, MROWS / 16), wave, 0, stream>>>(
        xc, c_proj_w, Cm, MROWS, DSTATE, DINNER);

    // 5) selective scan + skip (x*D) + SiLU(res) gating -> yg
    scan_kernel<<<(BB * DINNER) / 256, 256, 0, stream>>>(
        delta, Bm, Cm, xc, xr, A_log, Dp, yg);

    // 6) out = yg @ out_proj_w^T   (8192 x 512, K=1024)
    gemm_wmma_f32_32x32<<<dim3(DMODEL / 32, MROWS / 32), wave, 0, stream>>>(
        yg, out_proj_w, nullptr, out, MROWS, DMODEL, DINNER, 0);
}